// GravNetBlock_75222057222972
// MI455X (gfx1250) — compile-verified
//
#include <hip/hip_runtime.h>
#include <math.h>

typedef __attribute__((ext_vector_type(2))) float v2f;
typedef __attribute__((ext_vector_type(8))) float v8f;

#define KNN 40

// ---------------------------------------------------------------------------
// Zero a float region (accumulators in ws must be re-zeroed every launch).
// ---------------------------------------------------------------------------
__global__ void zero_f32(float* __restrict__ p, int n) {
    int i = blockIdx.x * blockDim.x + threadIdx.x;
    if (i < n) p[i] = 0.0f;
}

// ---------------------------------------------------------------------------
// C[M x 32] = act(A[M x Ka] @ W[Ka x 32] + bias), f32 WMMA 16x16x4.
// One wave computes a 16-row x 32-col tile (two 16x16 accumulators).
// Block = 128 threads = 4 waves -> 64 rows per block. M % 64 == 0, Ka % 4 == 0.
// ---------------------------------------------------------------------------
__global__ void __launch_bounds__(128)
gemm_n32_wmma(const float* __restrict__ A, const float* __restrict__ W,
              const float* __restrict__ bias, float* __restrict__ C,
              int M, int Ka, int applyElu)
{
    const int lane = threadIdx.x & 31;
    const int wave = threadIdx.x >> 5;
    const int half = lane >> 4;   // 0: lanes 0-15, 1: lanes 16-31
    const int lr   = lane & 15;
    const int row0 = blockIdx.x * 64 + wave * 16;
    if (row0 >= M) return;

    v8f acc0 = {0.f,0.f,0.f,0.f,0.f,0.f,0.f,0.f};
    v8f acc1 = {0.f,0.f,0.f,0.f,0.f,0.f,0.f,0.f};

    for (int kk = 0; kk < Ka; kk += 4) {
        const int k0 = kk + half * 2;
        // A fragment 16x4: lane holds A[row0+lr][k0], A[row0+lr][k0+1]
        const float* arow = A + (size_t)(row0 + lr) * Ka + k0;
        v2f a; a.x = arow[0]; a.y = arow[1];
        // B fragments 4x16: lane holds W[k0][n], W[k0+1][n]
        const float* w0 = W + (size_t)k0 * 32;
        v2f b0; b0.x = w0[lr];       b0.y = w0[32 + lr];
        v2f b1; b1.x = w0[16 + lr];  b1.y = w0[48 + lr];
        acc0 = __builtin_amdgcn_wmma_f32_16x16x4_f32(false, a, false, b0,
                                                     (short)0, acc0, false, false);
        acc1 = __builtin_amdgcn_wmma_f32_16x16x4_f32(false, a, false, b1,
                                                     (short)0, acc1, false, false);
    }

    const float bs0 = bias ? bias[lr]      : 0.0f;
    const float bs1 = bias ? bias[16 + lr] : 0.0f;
    #pragma unroll
    for (int v = 0; v < 8; ++v) {
        const int m = row0 + v + 8 * half;
        float c0 = acc0[v] + bs0;
        float c1 = acc1[v] + bs1;
        if (applyElu) {
            c0 = (c0 > 0.0f) ? c0 : (__expf(c0) - 1.0f);
            c1 = (c1 > 0.0f) ? c1 : (__expf(c1) - 1.0f);
        }
        C[(size_t)m * 32 + lr]      = c0;
        C[(size_t)m * 32 + 16 + lr] = c1;
    }
}

// ---------------------------------------------------------------------------
// Training-mode BatchNorm statistics per feature column (32 cols).
// grid = 32 blocks (one per feature), block = 256 threads.
// scale = g / sqrt(var + eps); shift = b - mu * scale
// ---------------------------------------------------------------------------
__global__ void bn_stats(const float* __restrict__ h,
                         const float* __restrict__ g, const float* __restrict__ b,
                         float* __restrict__ scale, float* __restrict__ shift, int N)
{
    const int c = blockIdx.x;
    __shared__ float ssum[256];
    __shared__ float ssq[256];
    float s = 0.0f, sq = 0.0f;
    for (int i = threadIdx.x; i < N; i += 256) {
        float v = h[(size_t)i * 32 + c];
        s += v; sq += v * v;
    }
    ssum[threadIdx.x] = s; ssq[threadIdx.x] = sq;
    __syncthreads();
    for (int o = 128; o > 0; o >>= 1) {
        if (threadIdx.x < o) {
            ssum[threadIdx.x] += ssum[threadIdx.x + o];
            ssq[threadIdx.x]  += ssq[threadIdx.x + o];
        }
        __syncthreads();
    }
    if (threadIdx.x == 0) {
        float mu  = ssum[0] / (float)N;
        float var = ssq[0] / (float)N - mu * mu;   // biased variance
        float sc  = g[c] * rsqrtf(var + 1e-5f);
        scale[c] = sc;
        shift[c] = b[c] - mu * sc;
    }
}

__global__ void bn_apply(const float* __restrict__ h,
                         const float* __restrict__ scale, const float* __restrict__ shift,
                         float* __restrict__ out, int total)
{
    int i = blockIdx.x * blockDim.x + threadIdx.x;
    if (i < total) {
        int c = i & 31;
        out[i] = h[i] * scale[c] + shift[c];
    }
}

// ---------------------------------------------------------------------------
// s_l = x_in @ lin_s_w  (32 x 3) -- too skinny for WMMA; plain VALU.
// ---------------------------------------------------------------------------
__global__ void s_proj(const float* __restrict__ xin, const float* __restrict__ ws,
                       float* __restrict__ sl, int N)
{
    int i = blockIdx.x * blockDim.x + threadIdx.x;
    if (i >= N) return;
    float a0 = 0.f, a1 = 0.f, a2 = 0.f;
    const float* xr = xin + (size_t)i * 32;
    #pragma unroll
    for (int c = 0; c < 32; ++c) {
        float x = xr[c];
        a0 += x * ws[c * 3 + 0];
        a1 += x * ws[c * 3 + 1];
        a2 += x * ws[c * 3 + 2];
    }
    sl[(size_t)i * 3 + 0] = a0;
    sl[(size_t)i * 3 + 1] = a1;
    sl[(size_t)i * 3 + 2] = a2;
}

// ---------------------------------------------------------------------------
// Brute-force kNN (K=40) in 3-D s-space, self-loops excluded.
// One thread per query; candidates staged through LDS in tiles of 256.
// Fully unrolled register insertion sort keeps best-40 in VGPRs.
// N % 256 == 0.
// ---------------------------------------------------------------------------
__global__ void __launch_bounds__(256)
knn_kernel(const float* __restrict__ s, int N,
           int* __restrict__ idx, float* __restrict__ esq, float* __restrict__ edist)
{
    __shared__ float sx[256], sy[256], sz[256];
    const int i = blockIdx.x * 256 + threadIdx.x;
    const float qx = s[(size_t)i * 3 + 0];
    const float qy = s[(size_t)i * 3 + 1];
    const float qz = s[(size_t)i * 3 + 2];

    float bd[KNN];
    int   bi[KNN];
    #pragma unroll
    for (int k = 0; k < KNN; ++k) { bd[k] = __builtin_inff(); bi[k] = -1; }

    for (int base = 0; base < N; base += 256) {
        int j = base + threadIdx.x;
        __syncthreads();
        sx[threadIdx.x] = s[(size_t)j * 3 + 0];
        sy[threadIdx.x] = s[(size_t)j * 3 + 1];
        sz[threadIdx.x] = s[(size_t)j * 3 + 2];
        __syncthreads();
        for (int t = 0; t < 256; ++t) {
            int j2 = base + t;
            float dx = qx - sx[t], dy = qy - sy[t], dz = qz - sz[t];
            float d2 = dx * dx + dy * dy + dz * dz;
            if (j2 == i) d2 = __builtin_inff();
            if (d2 < bd[KNN - 1]) {
                float dc = d2; int ic = j2;
                #pragma unroll
                for (int k = 0; k < KNN; ++k) {
                    if (dc < bd[k]) {
                        float tf = bd[k]; bd[k] = dc; dc = tf;
                        int   ti = bi[k]; bi[k] = ic; ic = ti;
                    }
                }
            }
        }
    }
    #pragma unroll
    for (int k = 0; k < KNN; ++k) {
        size_t e = (size_t)i * KNN + k;
        idx[e]   = bi[k];
        esq[e]   = bd[k];
        edist[e] = sqrtf(bd[k] + 1e-6f);
    }
}

// ---------------------------------------------------------------------------
// Edge reductions at dst = neighbor index.
// ---------------------------------------------------------------------------
__global__ void edge_accum1(const int* __restrict__ idx, const float* __restrict__ ed,
                            float* __restrict__ indeg, float* __restrict__ sumd, int E)
{
    int e = blockIdx.x * blockDim.x + threadIdx.x;
    if (e < E) {
        int d = idx[e];
        atomicAdd(&indeg[d], 1.0f);
        atomicAdd(&sumd[d], ed[e]);
    }
}

__global__ void node_loss1(const float* __restrict__ indeg, const float* __restrict__ sumd,
                           float* __restrict__ acc, int N)
{
    __shared__ float sh[256];
    int i = blockIdx.x * 256 + threadIdx.x;
    float v = 0.0f;
    if (i < N) {
        float deg = indeg[i];
        float av  = (deg > 0.0f) ? sumd[i] / fmaxf(deg, 1.0f) : 0.0f;
        float d   = av - 0.5f;
        v = d * d;
    }
    sh[threadIdx.x] = v;
    __syncthreads();
    for (int o = 128; o > 0; o >>= 1) {
        if (threadIdx.x < o) sh[threadIdx.x] += sh[threadIdx.x + o];
        __syncthreads();
    }
    if (threadIdx.x == 0) atomicAdd(acc, sh[0]);
}

__global__ void edge_accum2(const int* __restrict__ idx, const float* __restrict__ ed,
                            const float* __restrict__ sumd, float* __restrict__ degs2, int E)
{
    int e = blockIdx.x * blockDim.x + threadIdx.x;
    if (e < E) {
        int d = idx[e];
        float nrm = ed[e] / (sumd[d] + 1e-4f);
        atomicAdd(&degs2[d], nrm);
    }
}

__global__ void edge_loss2(const int* __restrict__ idx, const float* __restrict__ ed,
                           const float* __restrict__ sumd, const float* __restrict__ degs2,
                           float* __restrict__ acc, int E)
{
    __shared__ float sh[256];
    int e = blockIdx.x * 256 + threadIdx.x;
    float v = 0.0f;
    if (e < E) {
        int d = idx[e];
        float nrm = ed[e] / (sumd[d] + 1e-4f);
        float gn  = ed[e] / (degs2[d] + 1e-4f);
        float df  = nrm - gn;
        v = df * df;
    }
    sh[threadIdx.x] = v;
    __syncthreads();
    for (int o = 128; o > 0; o >>= 1) {
        if (threadIdx.x < o) sh[threadIdx.x] += sh[threadIdx.x + o];
        __syncthreads();
    }
    if (threadIdx.x == 0) atomicAdd(acc, sh[0]);
}

// ---------------------------------------------------------------------------
// msgs = h_l[idx] * edge_sq; mean & max over K. Builds cat96 = [mean|max|x_in].
// One thread per (node, feature).
// ---------------------------------------------------------------------------
__global__ void aggregate(const float* __restrict__ hl, const float* __restrict__ xin,
                          const int* __restrict__ idx, const float* __restrict__ esq,
                          float* __restrict__ cat96, int N)
{
    int t = blockIdx.x * blockDim.x + threadIdx.x;
    if (t >= N * 32) return;
    int i = t >> 5, p = t & 31;
    const int*   row = idx + (size_t)i * KNN;
    const float* er  = esq + (size_t)i * KNN;
    float sum = 0.0f, mx = -__builtin_inff();
    #pragma unroll 4
    for (int k = 0; k < KNN; ++k) {
        float v = hl[(size_t)row[k] * 32 + p] * er[k];
        sum += v;
        mx = fmaxf(mx, v);
    }
    cat96[(size_t)i * 96 + p]      = sum * (1.0f / (float)KNN);
    cat96[(size_t)i * 96 + 32 + p] = mx;
    cat96[(size_t)i * 96 + 64 + p] = xin[(size_t)i * 32 + p];
}

// cat68 = [xgn(32) | s_l(3) | x_in(32) | 0-pad(1)]
__global__ void build_cat68(const float* __restrict__ xgn, const float* __restrict__ sl,
                            const float* __restrict__ xin, float* __restrict__ cat, int N)
{
    int t = blockIdx.x * blockDim.x + threadIdx.x;
    if (t >= N * 68) return;
    int i = t / 68, c = t % 68;
    float v;
    if (c < 32)      v = xgn[(size_t)i * 32 + c];
    else if (c < 35) v = sl[(size_t)i * 3 + (c - 32)];
    else if (c < 67) v = xin[(size_t)i * 32 + (c - 35)];
    else             v = 0.0f;
    cat[t] = v;
}

// post_w1 (67x32) -> zero-padded (68x32)
__global__ void pad_w68(const float* __restrict__ w67, float* __restrict__ w68)
{
    int t = blockIdx.x * blockDim.x + threadIdx.x;
    if (t < 68 * 32) w68[t] = (t < 67 * 32) ? w67[t] : 0.0f;
}

__global__ void finalize_losses(const float* __restrict__ acc, float* __restrict__ out, int N)
{
    if (blockIdx.x == 0 && threadIdx.x == 0) {
        out[0] = 0.01f * acc[0] / (float)N;
        out[1] = 0.1f  * acc[1] / (float)(N * KNN);
    }
}

// ---------------------------------------------------------------------------
// Host-side orchestration
// ---------------------------------------------------------------------------
extern "C" void kernel_launch(void* const* d_in, const int* in_sizes, int n_in,
                              void* d_out, int out_size, void* d_ws, size_t ws_size,
                              hipStream_t stream)
{
    (void)n_in; (void)out_size; (void)ws_size;

    const float* x        = (const float*)d_in[0];   // [N,16]
    // d_in[1] original_coords, d_in[2] batch, d_in[3] step_count, d_in[4] num_layer: unused
    const float* pre_w1   = (const float*)d_in[5];   // [16,32]
    const float* pre_b1   = (const float*)d_in[6];
    const float* pre_w2   = (const float*)d_in[7];   // [32,32]
    const float* pre_b2   = (const float*)d_in[8];
    const float* bn1_g    = (const float*)d_in[9];
    const float* bn1_b    = (const float*)d_in[10];
    const float* lin_s_w  = (const float*)d_in[11];  // [32,3]
    const float* lin_h_w  = (const float*)d_in[12];  // [32,32]
    const float* lin_h_b  = (const float*)d_in[13];
    const float* lin_w    = (const float*)d_in[14];  // [96,32]
    const float* lin_b    = (const float*)d_in[15];
    const float* post_w1  = (const float*)d_in[16];  // [67,32]
    const float* post_b1  = (const float*)d_in[17];
    const float* post_w2  = (const float*)d_in[18];  // [32,32]
    const float* post_b2  = (const float*)d_in[19];
    const float* bn2_g    = (const float*)d_in[20];
    const float* bn2_b    = (const float*)d_in[21];

    float* out = (float*)d_out;
    const int N = in_sizes[0] / 16;   // 8192
    const int E = N * KNN;

    // workspace layout (floats)
    float* F = (float*)d_ws;
    size_t o = 0;
    float* t1    = F + o; o += (size_t)N * 32;
    float* t2    = F + o; o += (size_t)N * 32;
    float* xin   = F + o; o += (size_t)N * 32;
    float* hl    = F + o; o += (size_t)N * 32;
    float* sl    = F + o; o += (size_t)N * 3;
    int*   idxq  = (int*)(F + o); o += (size_t)N * KNN;
    float* esq   = F + o; o += (size_t)N * KNN;
    float* edst  = F + o; o += (size_t)N * KNN;
    float* indeg = F + o; o += (size_t)N;           // \  contiguous
    float* sumd  = F + o; o += (size_t)N;           //  | accumulator
    float* degs2 = F + o; o += (size_t)N;           //  | region
    float* lacc  = F + o; o += 2;                   // /  (3N+2 floats)
    float* cat96 = F + o; o += (size_t)N * 96;
    float* xgn   = F + o; o += (size_t)N * 32;
    float* cat68 = F + o; o += (size_t)N * 68;
    float* t3    = F + o; o += (size_t)N * 32;
    float* t4    = F + o; o += (size_t)N * 32;
    float* w68   = F + o; o += 68 * 32;
    float* sc1   = F + o; o += 32;
    float* sh1   = F + o; o += 32;
    float* sc2   = F + o; o += 32;
    float* sh2   = F + o; o += 32;

    const int gemmBlocks = N / 64;

    // 0. zero accumulators (ws is poisoned, never re-zeroed by harness)
    zero_f32<<<(3 * N + 2 + 255) / 256, 256, 0, stream>>>(indeg, 3 * N + 2);

    // 1-2. pre_gravnet MLP (elu)
    gemm_n32_wmma<<<gemmBlocks, 128, 0, stream>>>(x,  pre_w1, pre_b1, t1, N, 16, 1);
    gemm_n32_wmma<<<gemmBlocks, 128, 0, stream>>>(t1, pre_w2, pre_b2, t2, N, 32, 1);

    // 3. batchnorm 1 -> x_in
    bn_stats<<<32, 256, 0, stream>>>(t2, bn1_g, bn1_b, sc1, sh1, N);
    bn_apply<<<(N * 32) / 256, 256, 0, stream>>>(t2, sc1, sh1, xin, N * 32);

    // 4. GravNet projections
    gemm_n32_wmma<<<gemmBlocks, 128, 0, stream>>>(xin, lin_h_w, lin_h_b, hl, N, 32, 0);
    s_proj<<<N / 256, 256, 0, stream>>>(xin, lin_s_w, sl, N);

    // 5. kNN in s-space
    knn_kernel<<<N / 256, 256, 0, stream>>>(sl, N, idxq, esq, edst);

    // 6. edge-weight losses
    edge_accum1<<<(E + 255) / 256, 256, 0, stream>>>(idxq, edst, indeg, sumd, E);
    node_loss1<<<(N + 255) / 256, 256, 0, stream>>>(indeg, sumd, lacc + 0, N);
    edge_accum2<<<(E + 255) / 256, 256, 0, stream>>>(idxq, edst, sumd, degs2, E);
    edge_loss2<<<(E + 255) / 256, 256, 0, stream>>>(idxq, edst, sumd, degs2, lacc + 1, E);

    // 7. message passing + concat -> lin
    aggregate<<<(N * 32) / 256, 256, 0, stream>>>(hl, xin, idxq, esq, cat96, N);
    gemm_n32_wmma<<<gemmBlocks, 128, 0, stream>>>(cat96, lin_w, lin_b, xgn, N, 96, 0);

    // 8. post_gravnet MLP (67 padded to 68) + batchnorm 2 -> out
    pad_w68<<<(68 * 32 + 255) / 256, 256, 0, stream>>>(post_w1, w68);
    build_cat68<<<(N * 68 + 255) / 256, 256, 0, stream>>>(xgn, sl, xin, cat68, N);
    gemm_n32_wmma<<<gemmBlocks, 128, 0, stream>>>(cat68, w68, post_b1, t3, N, 68, 1);
    gemm_n32_wmma<<<gemmBlocks, 128, 0, stream>>>(t3, post_w2, post_b2, t4, N, 32, 1);
    bn_stats<<<32, 256, 0, stream>>>(t4, bn2_g, bn2_b, sc2, sh2, N);
    bn_apply<<<(N * 32) / 256, 256, 0, stream>>>(t4, sc2, sh2, out, N * 32);

    // 9. scalar losses appended after the [N,32] output
    finalize_losses<<<1, 32, 0, stream>>>(lacc, out + (size_t)N * 32, N);
}